// BiLSTM_CRF_1331439862245
// MI455X (gfx1250) — compile-verified
//
#include <hip/hip_runtime.h>
#include <math.h>

// ---------------- problem constants (match reference) ----------------
#define L_SEQ   4096
#define E_DIM   300
#define H2      256      // per-direction hidden
#define G4      1024     // 4*H2 gate rows
#define T_TAGS  16
#define START_TAG 14
#define STOP_TAG  15
#define NEGV   (-10000.0f)

typedef float v2f __attribute__((ext_vector_type(2)));
typedef float v8f __attribute__((ext_vector_type(8)));

// ---------------- workspace layout (bytes) ----------------
// xs      : L*E   f32   gathered embeddings
// zx      : 2*L*G4 f32  precomputed x@W_ih.T + b   (dir 0 = fwd, dir 1 = bwd)
// hf_all  : L*H2  f32   forward hidden states
// hb_all  : L*H2  f32   backward hidden states
// feats   : L*T   f32   emission scores
// hglob   : 2dir * 2parity * H2 f32   ping-pong h exchange buffers
// cnt     : 2 int        per-direction monotonic arrive counters
static constexpr size_t OFF_XS = 0;
static constexpr size_t OFF_ZX = OFF_XS + (size_t)L_SEQ * E_DIM * 4;
static constexpr size_t OFF_HF = OFF_ZX + (size_t)2 * L_SEQ * G4 * 4;
static constexpr size_t OFF_HB = OFF_HF + (size_t)L_SEQ * H2 * 4;
static constexpr size_t OFF_FE = OFF_HB + (size_t)L_SEQ * H2 * 4;
static constexpr size_t OFF_HG = OFF_FE + (size_t)L_SEQ * T_TAGS * 4;
static constexpr size_t OFF_CT = OFF_HG + (size_t)2 * 2 * H2 * 4;
// total ~47.2 MB

__device__ __forceinline__ float fsig(float x) { return 1.0f / (1.0f + expf(-x)); }

// ============ Kernel A: embedding gather + state init (resets sync each call) ============
__global__ void gather_init_kernel(const int* __restrict__ sent,
                                   const float* __restrict__ emb,
                                   const float* __restrict__ h0,
                                   float* __restrict__ xs,
                                   float* __restrict__ hglob,
                                   int* __restrict__ cnt) {
  int t = blockIdx.x;
  long row = (long)sent[t] * E_DIM;
  for (int e = threadIdx.x; e < E_DIM; e += blockDim.x)
    xs[(size_t)t * E_DIM + e] = emb[row + e];
  if (blockIdx.x == 0) {
    if (threadIdx.x < 2) cnt[threadIdx.x] = 0;            // re-arm per launch (graph-safe)
    for (int j = threadIdx.x; j < 2 * H2; j += blockDim.x) {
      int dir = j >> 8, jj = j & (H2 - 1);
      hglob[((size_t)dir * 2 + 0) * H2 + jj] = h0[j];     // parity-0 buffer = h0[dir]
    }
  }
}

// ============ Kernel B: Zx = xs @ W_ih^T + b for both directions (fp32 WMMA) ============
// One wave computes one 16x16 tile. M=4096, K=300 (75 k-steps of 4), N=1024, x2 dirs.
__global__ void input_gemm_kernel(const float* __restrict__ xs,
                                  const float* __restrict__ wihf,
                                  const float* __restrict__ bf,
                                  const float* __restrict__ wihb,
                                  const float* __restrict__ bb,
                                  float* __restrict__ zx) {
  int wave = (blockIdx.x * blockDim.x + threadIdx.x) >> 5;   // 32768 waves total
  int lane = threadIdx.x & 31;
  int dir  = wave >> 14;                                     // 16384 tiles per dir
  int tile = wave & 16383;
  int tm = tile >> 6;                                        // 256 m-tiles
  int tn = tile & 63;                                        // 64  n-tiles
  const float* W = dir ? wihb : wihf;                        // [1024,300] row-major
  const float* B = dir ? bb : bf;
  int m0 = tm << 4, n0 = tn << 4;
  int ml = lane & 15;
  int hi = lane >> 4;                                        // K half select (0: k0..1, 1: k2..3)
  v8f acc = {};
  for (int k0 = 0; k0 < E_DIM; k0 += 4) {
    int ka = k0 + (hi << 1);
    // A (16x4 f32): lanes 0-15 rows M, vgpr0/1 = K{0,1}; lanes 16-31 = K{2,3}
    v2f a = *(const v2f*)(xs + (size_t)(m0 + ml) * E_DIM + ka);
    // B (4x16 f32): B[k][n] = W[n][k]
    v2f b = *(const v2f*)(W + (size_t)(n0 + ml) * E_DIM + ka);
    acc = __builtin_amdgcn_wmma_f32_16x16x4_f32(false, a, false, b,
                                                (short)0, acc, false, false);
  }
  float bias = B[n0 + ml];
  float* zr = zx + (size_t)dir * L_SEQ * G4;
  #pragma unroll
  for (int r = 0; r < 8; ++r) {
    int row = m0 + r + (hi << 3);                            // C/D layout: vgpr r -> M=r / r+8
    zr[(size_t)row * G4 + n0 + ml] = acc[r] + bias;
  }
}

// ============ Kernel C: bidirectional LSTM recurrence ============
// 8 workgroups: blockIdx = dir*4 + wg. Each WGP holds its 256KB W_hh slice in LDS
// (hidden units j in [64*wg, 64*wg+64), all 4 gates). Per step: LDS matvec,
// gate nonlinearity, publish 64 h-values (agent scope), cluster barrier + atomic
// fallback, ping-pong h buffers.
__global__ void __launch_bounds__(256, 1)
bilstm_recurrent_kernel(const float* __restrict__ zx,
                        const float* __restrict__ whhf,
                        const float* __restrict__ whhb,
                        const float* __restrict__ c0,
                        float* __restrict__ hglob,
                        float* __restrict__ hf_all,
                        float* __restrict__ hb_all,
                        int* __restrict__ cnt,
                        int use_atomic_sync) {
  const int tid = threadIdx.x;
  const int dir = blockIdx.x >> 2;
  const int wg  = blockIdx.x & 3;
  extern __shared__ float smem[];
  float* sW = smem;                       // 256 rows * stride 260 (bank-conflict pad)
  float* sH = smem + 256 * 260;           // 256 current h
  float* sZ = sH + H2;                    // 256 gate pre-activations

  const float* Whh = dir ? whhb : whhf;   // [1024,256]
  // local row lr = g*64 + jj  -> global gate row G = g*256 + wg*64 + jj
  for (int lr = 0; lr < 256; ++lr) {
    int G = ((lr >> 6) << 8) + (wg << 6) + (lr & 63);
    sW[lr * 260 + tid] = Whh[(size_t)G * H2 + tid];         // coalesced, one row/iter
  }
  const int Gt = ((tid >> 6) << 8) + (wg << 6) + (tid & 63);  // this thread's gate row
  float c = (tid < 64) ? c0[dir * H2 + (wg << 6) + tid] : 0.0f;
  float* hg   = hglob + (size_t)dir * 2 * H2;
  float* hall = dir ? hb_all : hf_all;
  const float* zbase = zx + (size_t)dir * L_SEQ * G4;
  __syncthreads();

  int rp = 0;
  for (int s = 0; s < L_SEQ; ++s) {
    const int t = dir ? (L_SEQ - 1 - s) : s;
    // pull current h (written by all 4 WGPs) — device-scope loads stay L2-coherent
    sH[tid] = __hip_atomic_load(&hg[rp * H2 + tid], __ATOMIC_RELAXED,
                                __HIP_MEMORY_SCOPE_AGENT);
    if (s + 1 < L_SEQ) {
      const int tn = dir ? (t - 1) : (t + 1);
      __builtin_prefetch(&zbase[(size_t)tn * G4 + Gt], 0, 1);  // global_prefetch_b8
    }
    __syncthreads();
    // z = W_hh[row] . h  (+ precomputed x-part w/ bias)
    float acc = 0.0f;
    const float4* wr = (const float4*)(sW + tid * 260);
    const float4* hv = (const float4*)sH;
    #pragma unroll 8
    for (int i = 0; i < 64; ++i) {
      float4 w = wr[i], h = hv[i];
      acc += w.x * h.x + w.y * h.y + w.z * h.z + w.w * h.w;
    }
    sZ[tid] = acc + zbase[(size_t)t * G4 + Gt];
    __syncthreads();
    if (tid < 64) {
      float ig = fsig(sZ[tid]);
      float fg = fsig(sZ[64 + tid]);
      float gg = tanhf(sZ[128 + tid]);
      float og = fsig(sZ[192 + tid]);
      c = fg * c + ig * gg;
      float h = og * tanhf(c);
      int j = (wg << 6) + tid;
      __hip_atomic_store(&hg[(rp ^ 1) * H2 + j], h, __ATOMIC_RELAXED,
                         __HIP_MEMORY_SCOPE_AGENT);
      hall[(size_t)t * H2 + j] = h;
    }
    __threadfence();                 // publish h before arriving
    __syncthreads();
    // CDNA5 cluster split-barrier (s_barrier_signal/wait -3). NOP if the dispatch
    // is not cluster-formed, in which case the monotonic-counter sync below
    // provides the cross-WGP step barrier (8 co-resident WGs, deadlock-free).
    __builtin_amdgcn_s_cluster_barrier();
    if (use_atomic_sync && tid == 0) {
      __hip_atomic_fetch_add(&cnt[dir], 1, __ATOMIC_ACQ_REL,
                             __HIP_MEMORY_SCOPE_AGENT);
      const int target = 4 * (s + 1);
      while (__hip_atomic_load(&cnt[dir], __ATOMIC_ACQUIRE,
                               __HIP_MEMORY_SCOPE_AGENT) < target)
        __builtin_amdgcn_s_sleep(1);
    }
    __syncthreads();
    rp ^= 1;
  }
}

// ============ Kernel D: feats = [hf|hb] @ W_out^T + b_out (fp32 WMMA, N=16=T) ============
__global__ void proj_gemm_kernel(const float* __restrict__ hf_all,
                                 const float* __restrict__ hb_all,
                                 const float* __restrict__ Wout,
                                 const float* __restrict__ bout,
                                 float* __restrict__ feats) {
  int wave = (blockIdx.x * blockDim.x + threadIdx.x) >> 5;   // 256 waves, 16 rows each
  int lane = threadIdx.x & 31;
  int m0 = wave << 4;
  int ml = lane & 15, hi = lane >> 4;
  v8f acc = {};
  for (int k0 = 0; k0 < 2 * H2; k0 += 4) {
    int ka = k0 + (hi << 1);                                  // never straddles the 256 boundary
    const float* ap = (ka < H2) ? (hf_all + (size_t)(m0 + ml) * H2 + ka)
                                : (hb_all + (size_t)(m0 + ml) * H2 + (ka - H2));
    v2f a = *(const v2f*)ap;
    v2f b = *(const v2f*)(Wout + (size_t)ml * (2 * H2) + ka); // B[k][n] = Wout[n][k]
    acc = __builtin_amdgcn_wmma_f32_16x16x4_f32(false, a, false, b,
                                                (short)0, acc, false, false);
  }
  float bias = bout[ml];
  #pragma unroll
  for (int r = 0; r < 8; ++r)
    feats[(size_t)(m0 + r + (hi << 3)) * T_TAGS + ml] = acc[r] + bias;
}

// ============ Kernel E: Viterbi forward (wave32, shfl) + LDS backpointers + backtrack ====
__global__ void viterbi_kernel(const float* __restrict__ feats,
                               const float* __restrict__ trans,
                               float* __restrict__ out) {
  extern __shared__ unsigned char bp[];            // L*16 bytes of backpointers in LDS
  const int lane = threadIdx.x;
  const int n = lane & 15;                          // this lane's "next" tag
  float tr[T_TAGS];
  #pragma unroll
  for (int p = 0; p < T_TAGS; ++p) tr[p] = trans[n * T_TAGS + p];
  float fv = (n == START_TAG) ? 0.0f : NEGV;        // init forward var
  for (int t = 0; t < L_SEQ; ++t) {
    float feat = feats[t * T_TAGS + n];
    float m = -3.4e38f; int arg = 0;
    #pragma unroll
    for (int p = 0; p < T_TAGS; ++p) {
      float s = __shfl(fv, p, 32) + tr[p];          // fv[prev] + trans[next][prev]
      if (s > m) { m = s; arg = p; }                // strict > -> first max (argmax semantics)
    }
    fv = m + feat;
    if (lane < T_TAGS) bp[t * T_TAGS + n] = (unsigned char)arg;
  }
  float term = fv + trans[STOP_TAG * T_TAGS + n];
  float bm = -3.4e38f; int bi = 0;
  #pragma unroll
  for (int p = 0; p < T_TAGS; ++p) {
    float v = __shfl(term, p, 32);
    if (v > bm) { bm = v; bi = p; }
  }
  if (lane == 0) {
    out[0] = bm;                                    // score
    int tag = bi;
    for (int t = L_SEQ - 1; t >= 0; --t) {          // emit tag, then follow pointer
      out[1 + t] = (float)tag;
      tag = bp[t * T_TAGS + tag];
    }
  }
}

// ============ host launcher ============
extern "C" void kernel_launch(void* const* d_in, const int* in_sizes, int n_in,
                              void* d_out, int out_size, void* d_ws, size_t ws_size,
                              hipStream_t stream) {
  (void)in_sizes; (void)n_in; (void)out_size; (void)ws_size;
  const int*   sent  = (const int*)  d_in[0];
  const float* emb   = (const float*)d_in[1];
  const float* wihf  = (const float*)d_in[2];
  const float* whhf  = (const float*)d_in[3];
  const float* bf    = (const float*)d_in[4];
  const float* wihb  = (const float*)d_in[5];
  const float* whhb  = (const float*)d_in[6];
  const float* bb    = (const float*)d_in[7];
  const float* Wout  = (const float*)d_in[8];
  const float* bout  = (const float*)d_in[9];
  const float* trans = (const float*)d_in[10];
  const float* h0    = (const float*)d_in[11];
  const float* c0    = (const float*)d_in[12];
  float* out = (float*)d_out;

  char* ws = (char*)d_ws;
  float* xs     = (float*)(ws + OFF_XS);
  float* zx     = (float*)(ws + OFF_ZX);
  float* hf_all = (float*)(ws + OFF_HF);
  float* hb_all = (float*)(ws + OFF_HB);
  float* feats  = (float*)(ws + OFF_FE);
  float* hglob  = (float*)(ws + OFF_HG);
  int*   cnt    = (int*)  (ws + OFF_CT);

  gather_init_kernel<<<L_SEQ, 256, 0, stream>>>(sent, emb, h0, xs, hglob, cnt);

  // 2 dirs * (4096/16)*(1024/16) tiles = 32768 waves, 8 waves/block
  input_gemm_kernel<<<4096, 256, 0, stream>>>(xs, wihf, bf, wihb, bb, zx);

  const size_t smemC = (256 * 260 + 2 * H2) * sizeof(float);   // ~268 KB of the 320 KB LDS
  bilstm_recurrent_kernel<<<8, 256, smemC, stream>>>(zx, whhf, whhb, c0, hglob,
                                                     hf_all, hb_all, cnt,
                                                     /*use_atomic_sync=*/1);

  proj_gemm_kernel<<<32, 256, 0, stream>>>(hf_all, hb_all, Wout, bout, feats);

  viterbi_kernel<<<1, 32, (size_t)L_SEQ * T_TAGS, stream>>>(feats, trans, out);
}